// Discriminative_Frequency_Filter_Network_9363028706045
// MI455X (gfx1250) — compile-verified
//
#include <hip/hip_runtime.h>
#include <hip/hip_bf16.h>

#define C2      340
#define CIN     64
#define HIDDEN  170
#define COUT    64
#define WIDTH   256
#define NPIX    65536   // 256*256

typedef __attribute__((ext_vector_type(2))) float v2f;
typedef __attribute__((ext_vector_type(8))) float v8f;

__device__ __forceinline__ v8f wmma_f32_4(v2f a, v2f b, v8f c) {
  // (neg_a, A, neg_b, B, c_mod, C, reuse_a, reuse_b)
  return __builtin_amdgcn_wmma_f32_16x16x4_f32(false, a, false, b, (short)0, c,
                                               false, false);
}

// ---------------------------------------------------------------------------
// Kernel 0: h_c = irfft2(F_c)  (F real, (8,5) half-spectrum), then expand to
// the 64x64 circular-convolution matrix Mc[p][k] = h[(pp-xa)&7][(pq-xb)&7].
// Matches numpy irfft2: ifft over ky (e^{+i}), then irfft over kx.
// ---------------------------------------------------------------------------
__global__ void __launch_bounds__(64) k_build_M(const float* __restrict__ filt,
                                                float* __restrict__ Mc) {
  const int c = blockIdx.x;       // 0..339
  const int j = threadIdx.x;      // 0..63 -> (ny, m)
  __shared__ float hsh[64];
  const float* F = filt + c * 40; // (ky=8, kx=5)
  const int ny = j >> 3, m = j & 7;
  // exact cos/sin(2*pi*k/8) tables
  const float R2 = 0.70710678118654752440f;
  const float ct[8] = {1.f,  R2, 0.f, -R2, -1.f, -R2, 0.f,  R2};
  const float st[8] = {0.f,  R2, 1.f,  R2,  0.f, -R2, -1.f, -R2};

  float ur[5], ui[5];
  #pragma unroll
  for (int kx = 0; kx < 5; ++kx) {
    float re = 0.f, im = 0.f;
    #pragma unroll
    for (int ky = 0; ky < 8; ++ky) {
      int idx = (ky * ny) & 7;
      float f = F[ky * 5 + kx];
      re += f * ct[idx];
      im += f * st[idx];
    }
    ur[kx] = re * 0.125f;
    ui[kx] = im * 0.125f;
  }
  float hv = ur[0] + ((m & 1) ? -ur[4] : ur[4]);
  #pragma unroll
  for (int kx = 1; kx <= 3; ++kx) {
    int idx = (kx * m) & 7;
    hv += 2.f * (ur[kx] * ct[idx] - ui[kx] * st[idx]);
  }
  hsh[j] = hv * 0.125f;
  __syncthreads();

  float* Mrow = Mc + (size_t)c * 4096 + (size_t)j * 64;
  const int pp = ny, pq = m;
  #pragma unroll
  for (int k = 0; k < 64; ++k) {
    int xa = k >> 3, xb = k & 7;
    Mrow[k] = hsh[((pp - xa) & 7) * 8 + ((pq - xb) & 7)];
  }
}

// ---------------------------------------------------------------------------
// Kernel 1: z[b][o][n] = sum_c w_in[o][c] * x[b][c][n]   (M=340,K=64,N=65536/b)
// One wave computes one 16x16 output tile; A fragments resident, B streamed.
// Fragment layouts per CDNA5 ISA 7.12.2 (f32, 16x16x4).
// ---------------------------------------------------------------------------
__global__ void __launch_bounds__(256) k_gemm_in(const float* __restrict__ x,
                                                 const float* __restrict__ w_in,
                                                 float* __restrict__ z) {
  const int lane = threadIdx.x & 31;
  const int wave = threadIdx.x >> 5;
  const int lm = lane & 15;
  const int lh = lane >> 4;                 // half-wave select
  const int m0 = blockIdx.x * 16;           // 22 m-tiles (340 -> pad 352)
  const int b  = blockIdx.z;
  const int n  = (blockIdx.y * 8 + wave) * 16 + lm;

  // A fragments: w_in (C2 x CIN) row-major; K=64 -> 16 wmma k-steps
  v2f afrag[16];
  const int arow = m0 + lm;
  #pragma unroll
  for (int ks = 0; ks < 16; ++ks)
    #pragma unroll
    for (int v = 0; v < 2; ++v) {
      int k = ks * 4 + v + 2 * lh;
      afrag[ks][v] = (arow < C2) ? w_in[arow * CIN + k] : 0.0f;
    }

  const float* xb = x + (size_t)b * CIN * NPIX;
  v8f acc = {};
  #pragma unroll
  for (int ks = 0; ks < 16; ++ks) {
    v2f bfrag;
    #pragma unroll
    for (int v = 0; v < 2; ++v) {
      int k = ks * 4 + v + 2 * lh;
      bfrag[v] = xb[(size_t)k * NPIX + n];
    }
    acc = wmma_f32_4(afrag[ks], bfrag, acc);
  }

  float* zb = z + (size_t)b * C2 * NPIX;
  #pragma unroll
  for (int r = 0; r < 8; ++r) {
    int row = m0 + r + 8 * lh;
    if (row < C2) zb[(size_t)row * NPIX + n] = acc[r];
  }
}

// ---------------------------------------------------------------------------
// Kernel 2: per (channel,batch) apply Mc (64x64) to every 8x8 tile, IN PLACE.
// Y(64 x tiles) = Mc(64x64) * X(64 x tiles). Each lane-column owns one tile;
// all 64 pixels of a tile are gathered into B fragments before any store.
// ---------------------------------------------------------------------------
__global__ void __launch_bounds__(256) k_circ(float* __restrict__ z,
                                              const float* __restrict__ Mc) {
  const int lane = threadIdx.x & 31;
  const int wave = threadIdx.x >> 5;
  const int lm = lane & 15, lh = lane >> 4;
  const int c = blockIdx.x;
  const int b = blockIdx.z;
  const int t = (blockIdx.y * 8 + wave) * 16 + lm;   // tile 0..1023
  const int ty = t >> 5, tx = t & 31;                 // 32x32 tiles
  float* base = z + ((size_t)(b * C2 + c)) * NPIX + (size_t)ty * 8 * WIDTH + tx * 8;

  v2f bfrag[16];
  #pragma unroll
  for (int ks = 0; ks < 16; ++ks)
    #pragma unroll
    for (int v = 0; v < 2; ++v) {
      int k = ks * 4 + v + 2 * lh;                    // input pixel index 0..63
      bfrag[ks][v] = base[(k >> 3) * WIDTH + (k & 7)];
    }

  const float* M = Mc + (size_t)c * 4096;
  #pragma unroll
  for (int mt = 0; mt < 4; ++mt) {
    v8f acc = {};
    #pragma unroll
    for (int ks = 0; ks < 16; ++ks) {
      v2f afrag;
      #pragma unroll
      for (int v = 0; v < 2; ++v) {
        int k = ks * 4 + v + 2 * lh;
        afrag[v] = M[(mt * 16 + lm) * 64 + k];
      }
      acc = wmma_f32_4(afrag, bfrag[ks], acc);
    }
    #pragma unroll
    for (int r = 0; r < 8; ++r) {
      int p = mt * 16 + r + 8 * lh;                   // output pixel 0..63
      base[(p >> 3) * WIDTH + (p & 7)] = acc[r];
    }
  }
}

// ---------------------------------------------------------------------------
// Kernel 3: depthwise 3x3 (cross-correlation, SAME) + exact GELU gating.
// g[b][c] = gelu(dw(z[b][c])) * dw(z[b][c+170])
// ---------------------------------------------------------------------------
__global__ void __launch_bounds__(256) k_dwglu(const float* __restrict__ z,
                                               const float* __restrict__ dwk,
                                               float* __restrict__ g) {
  const int c = blockIdx.x;           // 0..169
  const int b = blockIdx.z;
  const int y0 = blockIdx.y * 16;
  const int xc = threadIdx.x;         // column 0..255
  const float* z1 = z + ((size_t)(b * C2 + c)) * NPIX;
  const float* z2 = z + ((size_t)(b * C2 + c + HIDDEN)) * NPIX;
  float* go = g + ((size_t)(b * HIDDEN + c)) * NPIX;

  float w1[9], w2[9];
  #pragma unroll
  for (int i = 0; i < 9; ++i) {
    w1[i] = dwk[c * 9 + i];
    w2[i] = dwk[(c + HIDDEN) * 9 + i];
  }

  for (int yy = 0; yy < 16; ++yy) {
    int y = y0 + yy;
    float s1 = 0.f, s2 = 0.f;
    #pragma unroll
    for (int dy = 0; dy < 3; ++dy) {
      int ys = y + dy - 1;
      if (ys < 0 || ys >= WIDTH) continue;
      #pragma unroll
      for (int dx = 0; dx < 3; ++dx) {
        int xs = xc + dx - 1;
        if (xs < 0 || xs >= WIDTH) continue;
        s1 += w1[dy * 3 + dx] * z1[ys * WIDTH + xs];
        s2 += w2[dy * 3 + dx] * z2[ys * WIDTH + xs];
      }
    }
    float ge = 0.5f * s1 * (1.0f + erff(s1 * 0.70710678118654752f));
    go[y * WIDTH + xc] = ge * s2;
  }
}

// ---------------------------------------------------------------------------
// Kernel 4: out[b][o][n] = sum_c w_out[o][c] * g[b][c][n]  (M=64,K=170->172)
// ---------------------------------------------------------------------------
__global__ void __launch_bounds__(256) k_gemm_out(const float* __restrict__ g,
                                                  const float* __restrict__ w_out,
                                                  float* __restrict__ out) {
  const int lane = threadIdx.x & 31;
  const int wave = threadIdx.x >> 5;
  const int lm = lane & 15, lh = lane >> 4;
  const int m0 = blockIdx.x * 16;     // 4 m-tiles (M=64)
  const int b  = blockIdx.z;
  const int n  = (blockIdx.y * 8 + wave) * 16 + lm;

  const float* gb = g + (size_t)b * HIDDEN * NPIX;
  v8f acc = {};
  #pragma unroll
  for (int ks = 0; ks < 43; ++ks) {   // K = 170 padded to 172
    v2f afrag, bfrag;
    #pragma unroll
    for (int v = 0; v < 2; ++v) {
      int k = ks * 4 + v + 2 * lh;
      bool ok = (k < HIDDEN);
      afrag[v] = ok ? w_out[(m0 + lm) * HIDDEN + k] : 0.0f;
      bfrag[v] = ok ? gb[(size_t)k * NPIX + n] : 0.0f;
    }
    acc = wmma_f32_4(afrag, bfrag, acc);
  }

  float* ob = out + (size_t)b * COUT * NPIX;
  #pragma unroll
  for (int r = 0; r < 8; ++r)
    ob[(size_t)(m0 + r + 8 * lh) * NPIX + n] = acc[r];
}

// ---------------------------------------------------------------------------
extern "C" void kernel_launch(void* const* d_in, const int* in_sizes, int n_in,
                              void* d_out, int out_size, void* d_ws, size_t ws_size,
                              hipStream_t stream) {
  const float* x     = (const float*)d_in[0];  // (4,64,256,256)
  const float* w_in  = (const float*)d_in[1];  // (340,64)
  const float* dw_k  = (const float*)d_in[2];  // (340,1,3,3)
  const float* filt  = (const float*)d_in[3];  // (340,1,1,8,5)
  const float* w_out = (const float*)d_in[4];  // (64,170)
  float* out = (float*)d_out;                  // (4,64,256,256)

  float* z  = (float*)d_ws;                    // (4,340,256,256) f32
  float* g  = z + (size_t)4 * C2 * NPIX;       // (4,170,256,256) f32
  float* Mc = g + (size_t)4 * HIDDEN * NPIX;   // (340,64,64)     f32

  k_build_M<<<dim3(C2, 1, 1),   dim3(64),  0, stream>>>(filt, Mc);
  k_gemm_in<<<dim3(22, 512, 4), dim3(256), 0, stream>>>(x, w_in, z);
  k_circ   <<<dim3(C2, 8, 4),   dim3(256), 0, stream>>>(z, Mc);
  k_dwglu  <<<dim3(HIDDEN, 16, 4), dim3(256), 0, stream>>>(z, dw_k, g);
  k_gemm_out<<<dim3(4, 512, 4), dim3(256), 0, stream>>>(g, w_out, out);
}